// GraphSynergy_54631984005171
// MI455X (gfx1250) — compile-verified
//
#include <hip/hip_runtime.h>
#include <hip/hip_bf16.h>
#include <math.h>
#include <stdint.h>

typedef float v2f __attribute__((ext_vector_type(2)));
typedef float v8f __attribute__((ext_vector_type(8)));

// ---------------------------------------------------------------------------
// Kernel 1: per-image CNN (conv1 3x3 1->4 + relu + pool2, conv2 3x3 4->8 +
// relu + pool2). One block per image, features (8192 f32) written to ws F.
// Dynamic LDS = exactly 64KB (pooled conv1 output); weights read via WGP$.
// ---------------------------------------------------------------------------
__global__ __launch_bounds__(256) void cnn_kernel(
    const float* __restrict__ cell_pre, const float* __restrict__ drug_pre,
    const float* __restrict__ w1g, const float* __restrict__ b1g,
    const float* __restrict__ w2g, const float* __restrict__ b2g,
    float* __restrict__ F)
{
    extern __shared__ float s_p1[];  // 4*64*64 = 16384 floats = 64KB

    const int img = blockIdx.x;
    const int t   = threadIdx.x;
    const float* in = (img < 76) ? (cell_pre + (size_t)img * 16384)
                                 : (drug_pre + (size_t)(img - 76) * 16384);

    // conv1 + relu + maxpool2 -> s_p1[c*4096 + ph*64 + pw], 4x64x64
    for (int o = t; o < 16384; o += 256) {
        const int c = o >> 12, rem = o & 4095, ph = rem >> 6, pw = rem & 63;
        const float bias = b1g[c];
        float mx = 0.f;  // relu >= 0
        for (int sy = 0; sy < 2; ++sy)
            for (int sx = 0; sx < 2; ++sx) {
                const int y = 2 * ph + sy, x = 2 * pw + sx;
                float acc = bias;
                for (int ky = 0; ky < 3; ++ky) {
                    const int iy = y + ky - 1;
                    if (iy < 0 || iy > 127) continue;
                    for (int kx = 0; kx < 3; ++kx) {
                        const int ix = x + kx - 1;
                        if (ix < 0 || ix > 127) continue;
                        acc += in[iy * 128 + ix] * w1g[c * 9 + ky * 3 + kx];
                    }
                }
                mx = fmaxf(mx, fmaxf(acc, 0.f));
            }
        s_p1[o] = mx;
    }
    __syncthreads();

    // conv2 + relu + maxpool2 -> F[img][c*1024 + ph*32 + pw], 8x32x32
    float* Fo = F + (size_t)img * 8192;
    for (int o = t; o < 8192; o += 256) {
        const int c = o >> 10, rem = o & 1023, ph = rem >> 5, pw = rem & 31;
        const float bias = b2g[c];
        float mx = 0.f;
        for (int sy = 0; sy < 2; ++sy)
            for (int sx = 0; sx < 2; ++sx) {
                const int y = 2 * ph + sy, x = 2 * pw + sx;
                float acc = bias;
                for (int i = 0; i < 4; ++i)
                    for (int ky = 0; ky < 3; ++ky) {
                        const int iy = y + ky - 1;
                        if (iy < 0 || iy > 63) continue;
                        for (int kx = 0; kx < 3; ++kx) {
                            const int ix = x + kx - 1;
                            if (ix < 0 || ix > 63) continue;
                            acc += s_p1[i * 4096 + iy * 64 + ix] *
                                   w2g[((c * 4 + i) * 3 + ky) * 3 + kx];
                        }
                    }
                mx = fmaxf(mx, fmaxf(acc, 0.f));
            }
        Fo[o] = mx;
    }
}

// ---------------------------------------------------------------------------
// Kernel 2: FC GEMM via f32 WMMA.  C[840x64] = F[840x8192] * out_w^T[8192x64]
// One wave per 16x16 C tile; block = 4 waves = the 4 N-tiles of one M-tile.
// A 16x4 f32 layout: lanes0-15 hold K={0,1}, lanes16-31 hold K={2,3}.
// ---------------------------------------------------------------------------
__global__ __launch_bounds__(128) void fc_wmma_kernel(
    const float* __restrict__ F, const float* __restrict__ out_w,
    float* __restrict__ Cfc)
{
    const int lane    = threadIdx.x & 31;
    const int nt      = threadIdx.x >> 5;   // wave id = N tile (0..3)
    const int mt      = blockIdx.x;         // 0..52
    const int half    = lane >> 4;          // K group 0/1
    const int laneMod = lane & 15;          // M (for A) / N (for B)

    int rowA = mt * 16 + laneMod;
    if (rowA > 839) rowA = 839;             // clamp; masked at store
    const float* aPtr = F + (size_t)rowA * 8192 + 2 * half;
    const float* bPtr = out_w + (size_t)(nt * 16 + laneMod) * 8192 + 2 * half;

    v8f acc = {};
#pragma unroll 8
    for (int k = 0; k < 8192; k += 4) {
        if ((k & 255) == 0) {
            __builtin_prefetch(aPtr + k + 1024, 0, 3);
            __builtin_prefetch(bPtr + k + 1024, 0, 3);
        }
        v2f a = *(const v2f*)(aPtr + k);
        v2f b = *(const v2f*)(bPtr + k);
        acc = __builtin_amdgcn_wmma_f32_16x16x4_f32(
            false, a, false, b, (short)0, acc, false, false);
    }

#pragma unroll
    for (int j = 0; j < 8; ++j) {
        const int row = mt * 16 + j + 8 * half;
        if (row < 840) Cfc[row * 64 + nt * 16 + laneMod] = acc[j];
    }
}

// ---------------------------------------------------------------------------
// Kernel 3: per-group (cells rows [0,76), drugs rows [76,840)) column-wise
// batch-norm with ddof=1.  grid=2 blocks x 64 threads (one per column).
// ---------------------------------------------------------------------------
__global__ __launch_bounds__(64) void norm_kernel(
    const float* __restrict__ Cfc, const float* __restrict__ out_b,
    float* __restrict__ item)
{
    const int g = blockIdx.x, j = threadIdx.x;
    const int r0 = g ? 76 : 0;
    const int n  = g ? 764 : 76;
    const float b = out_b[j];
    float s = 0.f;
    for (int r = 0; r < n; ++r) s += Cfc[(r0 + r) * 64 + j] + b;
    const float mean = s / (float)n;
    float v = 0.f;
    for (int r = 0; r < n; ++r) {
        const float d = Cfc[(r0 + r) * 64 + j] + b - mean;
        v += d * d;
    }
    const float sd = sqrtf(v / (float)(n - 1));
    for (int r = 0; r < n; ++r)
        item[(r0 + r) * 64 + j] = (Cfc[(r0 + r) * 64 + j] + b - mean) / sd;
}

// ---------------------------------------------------------------------------
// Kernel 4: interact (attention over 128 gathered protein rows per hop) +
// agg GEMV.  One block (128 threads) per node n in [0,840).
// The 128x64 f32 gather is staged into LDS once per hop with CDNA5
// global_load_async_to_lds_b128 (ASYNCcnt), then reused by both the score
// pass and the attention-weighted sum.  Rows padded to 68 floats (272B,
// 16B-aligned) to avoid LDS bank conflicts in the score pass.
// ---------------------------------------------------------------------------
__global__ __launch_bounds__(128) void interact_kernel(
    const int* __restrict__ cell_nbr, const int* __restrict__ drug_nbr,
    const float* __restrict__ protein, const float* __restrict__ item,
    const float* __restrict__ agg_w, const float* __restrict__ agg_b,
    float* __restrict__ emb)
{
    __shared__ float s_nbr[128 * 68];   // ~34KB, padded rows
    __shared__ float s_item[64];
    __shared__ float s_attn[128];
    __shared__ float s_red[128];
    __shared__ float s_hop[128];

    const int n = blockIdx.x, t = threadIdx.x;
    const int* nbr = (n < 76) ? (cell_nbr + n * 256) : (drug_nbr + (n - 76) * 256);
    if (t < 64) s_item[t] = item[n * 64 + t];

    for (int h = 0; h < 2; ++h) {
        const int* nh = nbr + h * 128;
        // async gather: thread t copies protein row nh[t] (256B) into LDS
        {
            const float* g = protein + (size_t)nh[t] * 64;
            const unsigned ldsoff =
                (unsigned)(uintptr_t)(const void*)(&s_nbr[t * 68]);
#pragma unroll
            for (int i = 0; i < 16; ++i) {
                asm volatile("global_load_async_to_lds_b128 %0, %1, off"
                             :: "v"(ldsoff + 16u * i), "v"(g + 4 * i)
                             : "memory");
            }
        }
        asm volatile("s_wait_asynccnt 0x0" ::: "memory");
        __syncthreads();

        // pass 1: attention scores from LDS
        float sc = 0.f;
        for (int d = 0; d < 64; ++d) sc += s_nbr[t * 68 + d] * s_item[d];
        s_red[t] = sc;
        __syncthreads();
        for (int off = 64; off > 0; off >>= 1) {
            if (t < off) s_red[t] = fmaxf(s_red[t], s_red[t + off]);
            __syncthreads();
        }
        const float mx = s_red[0];
        __syncthreads();
        const float ex = expf(sc - mx);
        s_attn[t] = ex;
        s_red[t]  = ex;
        __syncthreads();
        for (int off = 64; off > 0; off >>= 1) {
            if (t < off) s_red[t] += s_red[t + off];
            __syncthreads();
        }
        const float inv = 1.f / s_red[0];
        __syncthreads();

        // pass 2: attention-weighted sum from LDS
        if (t < 64) {
            float acc = 0.f;
            for (int k = 0; k < 128; ++k)
                acc += s_attn[k] * s_nbr[k * 68 + t];
            s_hop[h * 64 + t] = acc * inv;
        }
        __syncthreads();   // protects s_nbr/s_red reuse in next hop
    }

    if (t < 64) {
        float acc = agg_b[t];
        const float* w = agg_w + t * 128;
        for (int j = 0; j < 128; ++j) acc += w[j] * s_hop[j];
        emb[n * 64 + t] = acc;
    }
}

// ---------------------------------------------------------------------------
// Kernel 5: per-sample score + item_reg.  One block (64 threads) per sample.
// score = sum_m(combined[m]*ce[m]) - dot(d1e,d2e)
// ---------------------------------------------------------------------------
__global__ __launch_bounds__(64) void score_kernel(
    const int* __restrict__ data, const float* __restrict__ emb,
    const float* __restrict__ comb_w, float* __restrict__ score,
    float* __restrict__ itemreg)
{
    __shared__ float s1[64], s2[64], sce[64], sred[64];
    const int i = blockIdx.x, m = threadIdx.x;
    const int c  = data[i * 3 + 0];
    const int d1 = data[i * 3 + 1];
    const int d2 = data[i * 3 + 2];
    sce[m] = emb[c * 64 + m];
    s1[m]  = emb[(76 + d1) * 64 + m];
    s2[m]  = emb[(76 + d2) * 64 + m];
    __syncthreads();

    float comb = 0.f;
    const float* w = comb_w + m * 128;
    for (int j = 0; j < 64; ++j) comb += w[j] * s1[j];
    for (int j = 0; j < 64; ++j) comb += w[64 + j] * s2[j];

    sred[m] = comb * sce[m] - s1[m] * s2[m];
    __syncthreads();
    for (int off = 32; off > 0; off >>= 1) {
        if (m < off) sred[m] += sred[m + off];
        __syncthreads();
    }
    if (m == 0) score[i] = sred[0];
    __syncthreads();

    sred[m] = sce[m] * sce[m] + s1[m] * s1[m] + s2[m] * s2[m];
    __syncthreads();
    for (int off = 32; off > 0; off >>= 1) {
        if (m < off) sred[m] += sred[m + off];
        __syncthreads();
    }
    if (m == 0) itemreg[i] = 0.5f * sred[0];
}

// ---------------------------------------------------------------------------
// Kernel 6: deterministic final reduction: sum(itemreg) + node_reg (only
// samples hop=0,1 contribute, matching the reference's cnl[hop] indexing),
// emb_loss = 1e-6*(item+node)/4096 -> d_out[4096].
// ---------------------------------------------------------------------------
__global__ __launch_bounds__(256) void finalize_kernel(
    const int* __restrict__ data,
    const int* __restrict__ cell_nbr, const int* __restrict__ drug_nbr,
    const float* __restrict__ protein, const float* __restrict__ itemreg,
    float* __restrict__ out_loss)
{
    __shared__ float sred[256];
    const int t = threadIdx.x;
    float acc = 0.f;
    for (int i = t; i < 4096; i += 256) acc += itemreg[i];

    float node = 0.f;
    for (int hop = 0; hop < 2; ++hop) {
        const int c  = data[hop * 3 + 0];
        const int d1 = data[hop * 3 + 1];
        const int d2 = data[hop * 3 + 2];
        const int* lists[3] = { cell_nbr + c * 256,
                                drug_nbr + d1 * 256,
                                drug_nbr + d2 * 256 };
        for (int L = 0; L < 3; ++L) {
            const int* lst = lists[L];
            float s = 0.f;
            for (int idx = t; idx < 256 * 64; idx += 256) {
                const int k = idx >> 6, d = idx & 63;
                const float v = protein[(size_t)lst[k] * 64 + d];
                s += v * v;
            }
            node += 0.5f * s;
        }
    }
    sred[t] = acc + node;
    __syncthreads();
    for (int off = 128; off > 0; off >>= 1) {
        if (t < off) sred[t] += sred[t + off];
        __syncthreads();
    }
    if (t == 0) out_loss[0] = 1e-6f * sred[0] / 4096.f;
}

// ---------------------------------------------------------------------------
extern "C" void kernel_launch(void* const* d_in, const int* in_sizes, int n_in,
                              void* d_out, int out_size, void* d_ws, size_t ws_size,
                              hipStream_t stream)
{
    const int*   data     = (const int*)  d_in[0];
    const int*   cell_nbr = (const int*)  d_in[1];
    const int*   drug_nbr = (const int*)  d_in[2];
    const float* protein  = (const float*)d_in[3];
    const float* cell_pre = (const float*)d_in[4];
    const float* drug_pre = (const float*)d_in[5];
    const float* w1       = (const float*)d_in[6];
    const float* b1       = (const float*)d_in[7];
    const float* w2       = (const float*)d_in[8];
    const float* b2       = (const float*)d_in[9];
    const float* out_w    = (const float*)d_in[10];
    const float* out_b    = (const float*)d_in[11];
    const float* agg_w    = (const float*)d_in[12];
    const float* agg_b    = (const float*)d_in[13];
    const float* comb_w   = (const float*)d_in[14];

    float* F       = (float*)d_ws;          // 840*8192
    float* Cfc     = F + 840 * 8192;        // 840*64
    float* item    = Cfc + 840 * 64;        // 840*64
    float* emb     = item + 840 * 64;       // 840*64
    float* itemreg = emb + 840 * 64;        // 4096
    float* score   = (float*)d_out;         // 4096
    float* loss    = score + 4096;          // 1

    const size_t shbytes = 16384 * sizeof(float);   // 64KB
    cnn_kernel<<<840, 256, shbytes, stream>>>(cell_pre, drug_pre, w1, b1, w2, b2, F);
    fc_wmma_kernel<<<53, 128, 0, stream>>>(F, out_w, Cfc);
    norm_kernel<<<2, 64, 0, stream>>>(Cfc, out_b, item);
    interact_kernel<<<840, 128, 0, stream>>>(cell_nbr, drug_nbr, protein, item,
                                             agg_w, agg_b, emb);
    score_kernel<<<4096, 64, 0, stream>>>(data, emb, comb_w, score, itemreg);
    finalize_kernel<<<1, 256, 0, stream>>>(data, cell_nbr, drug_nbr, protein,
                                           itemreg, loss);
}